// GAT_35175782154374
// MI455X (gfx1250) — compile-verified
//
#include <hip/hip_runtime.h>
#include <math.h>

#define NEG_SLOPE 0.2f
#define FDIM 256
#define HEADS 4
#define HDIM 64
#define KTILE 32
#define LDT 34   // padded LDS row stride (floats): even (b64-aligned) + conflict-free

typedef float v2f __attribute__((ext_vector_type(2)));
typedef float v8f __attribute__((ext_vector_type(8)));

// ---- order-preserving float<->uint key for atomic max on f32 values ----
__device__ __forceinline__ unsigned f2key(float f) {
  unsigned u = __float_as_uint(f);
  unsigned mask = (unsigned)(((int)u) >> 31) | 0x80000000u;
  return u ^ mask;
}
__device__ __forceinline__ float key2f(unsigned k) {
  unsigned mask = (k & 0x80000000u) ? 0x80000000u : 0xFFFFFFFFu;
  return __uint_as_float(k ^ mask);
}
__device__ __forceinline__ void atomicAddF32(float* p, float v) {
  __hip_atomic_fetch_add(p, v, __ATOMIC_RELAXED, __HIP_MEMORY_SCOPE_AGENT);
}

// =====================================================================
// OUT = H (N x 256) * W (256 x 256), f32 WMMA 16x16x4.
// Block = 128 threads (4 waves) -> 64x64 output tile.
// Both operands staged in LDS per 32-wide K-tile; B stored transposed so
// every WMMA operand is one conflict-free ds_load_b64.
// Wave w computes rows [w*16, w*16+16) of the tile, all 64 columns.
// =====================================================================
__global__ __launch_bounds__(128) void gemm_f32_wmma(
    const float* __restrict__ H, const float* __restrict__ W,
    float* __restrict__ OUT, int N) {
  __shared__ float As[64 * LDT];   // As[r][k], r in 0..63, k in 0..31
  __shared__ float Bt[64 * LDT];   // Bt[n][k], n in 0..63, k in 0..31

  const int row0 = blockIdx.x * 64;
  const int col0 = blockIdx.y * 64;
  const int t = threadIdx.x;
  const int wave = t >> 5, lane = t & 31;
  const int hi = lane >> 4;        // half-wave selects K pair
  const int m  = lane & 15;

  v8f acc0 = {}, acc1 = {}, acc2 = {}, acc3 = {};

  for (int kt = 0; kt < FDIM; kt += KTILE) {
    __syncthreads();   // previous tile's compute done before restaging
    // stage A: 64x32 (coalesced 32-wide row segments)
    for (int i = t; i < 64 * KTILE; i += 128) {
      const int r = i >> 5, c = i & 31;
      int rr = row0 + r; if (rr > N - 1) rr = N - 1;
      As[r * LDT + c] = H[(size_t)rr * FDIM + kt + c];
    }
    // stage B transposed: Bt[n][k] = W[kt+k][col0+n] (coalesced 64-wide rows)
    for (int i = t; i < 64 * KTILE; i += 128) {
      const int k = i >> 6, n = i & 63;
      Bt[n * LDT + k] = W[(size_t)(kt + k) * FDIM + col0 + n];
    }
    __syncthreads();

    #pragma unroll
    for (int kk = 0; kk < KTILE; kk += 4) {
      const int ka = kk + 2 * hi;  // even -> 8B-aligned LDS b64 reads
      v2f a  = *(const v2f*)&As[(wave * 16 + m) * LDT + ka];
      v2f b0 = *(const v2f*)&Bt[(m)      * LDT + ka];
      v2f b1 = *(const v2f*)&Bt[(16 + m) * LDT + ka];
      v2f b2 = *(const v2f*)&Bt[(32 + m) * LDT + ka];
      v2f b3 = *(const v2f*)&Bt[(48 + m) * LDT + ka];
      acc0 = __builtin_amdgcn_wmma_f32_16x16x4_f32(false, a, false, b0, (short)0, acc0, false, false);
      acc1 = __builtin_amdgcn_wmma_f32_16x16x4_f32(false, a, false, b1, (short)0, acc1, false, false);
      acc2 = __builtin_amdgcn_wmma_f32_16x16x4_f32(false, a, false, b2, (short)0, acc2, false, false);
      acc3 = __builtin_amdgcn_wmma_f32_16x16x4_f32(false, a, false, b3, (short)0, acc3, false, false);
    }
  }

  // C/D layout: vgpr r -> tile row (r + 8*hi), lane m -> column within 16
  #pragma unroll
  for (int r = 0; r < 8; ++r) {
    const int row = row0 + wave * 16 + r + 8 * hi;
    if (row < N) {
      float* o = OUT + (size_t)row * FDIM + col0 + m;
      o[0]  = acc0[r];
      o[16] = acc1[r];
      o[32] = acc2[r];
      o[48] = acc3[r];
    }
  }
}

// =====================================================================
// el[n][h] = sum_d feat[n][h*64+d] * al[h][d]; er likewise.
// Also resets emax-key (-inf) and denom. One wave per node.
// =====================================================================
__global__ __launch_bounds__(256) void node_attn(
    const float* __restrict__ feat, const float* __restrict__ al,
    const float* __restrict__ ar, float* __restrict__ el,
    float* __restrict__ er, unsigned* __restrict__ emaxk,
    float* __restrict__ denom, int N) {
  const int lane = threadIdx.x & 31;
  const int n = blockIdx.x * 8 + (threadIdx.x >> 5);
  if (n >= N) return;
  const float* f = feat + (size_t)n * FDIM;
  #pragma unroll
  for (int h = 0; h < HEADS; ++h) {
    const int b = h * HDIM;
    float p = f[b + lane] * al[b + lane] + f[b + 32 + lane] * al[b + 32 + lane];
    float q = f[b + lane] * ar[b + lane] + f[b + 32 + lane] * ar[b + 32 + lane];
    #pragma unroll
    for (int off = 16; off > 0; off >>= 1) {
      p += __shfl_xor(p, off, 32);
      q += __shfl_xor(q, off, 32);
    }
    if (lane == 0) {
      el[n * HEADS + h] = p;
      er[n * HEADS + h] = q;
      emaxk[n * HEADS + h] = 0x007FFFFFu;  // f2key(-inf)
      denom[n * HEADS + h] = 0.0f;
    }
  }
}

// rst[i] = residual ? h[i] : 0   (float4 granularity)
__global__ __launch_bounds__(256) void init_acc(
    const float* __restrict__ hres, float* __restrict__ rst, int total4) {
  const int i = blockIdx.x * blockDim.x + threadIdx.x;
  if (i < total4) {
    float4 v = hres ? ((const float4*)hres)[i] : make_float4(0.f, 0.f, 0.f, 0.f);
    ((float4*)rst)[i] = v;
  }
}

// segment max of leaky_relu(el[src]+er[dst]) via u32-key atomic max
__global__ __launch_bounds__(256) void edge_max(
    const int* __restrict__ src, const int* __restrict__ dst,
    const float* __restrict__ el, const float* __restrict__ er,
    unsigned* __restrict__ emaxk, int E) {
  const int e = blockIdx.x * blockDim.x + threadIdx.x;
  if (e >= E) return;
  const int s = src[e], d = dst[e];
  const float4 a = ((const float4*)el)[s];
  const float4 b = ((const float4*)er)[d];
  float v[4] = {a.x + b.x, a.y + b.y, a.z + b.z, a.w + b.w};
  #pragma unroll
  for (int h = 0; h < 4; ++h) {
    float x = v[h];
    x = x > 0.f ? x : NEG_SLOPE * x;
    atomicMax(&emaxk[d * 4 + h], f2key(x));
  }
}

// ex = exp(e - emax[dst]); segment sum into denom; cache ex per edge
__global__ __launch_bounds__(256) void edge_sum(
    const int* __restrict__ src, const int* __restrict__ dst,
    const float* __restrict__ el, const float* __restrict__ er,
    const unsigned* __restrict__ emaxk, float* __restrict__ denom,
    float* __restrict__ exbuf, int E) {
  const int e = blockIdx.x * blockDim.x + threadIdx.x;
  if (e >= E) return;
  const int s = src[e], d = dst[e];
  const float4 a = ((const float4*)el)[s];
  const float4 b = ((const float4*)er)[d];
  float v[4] = {a.x + b.x, a.y + b.y, a.z + b.z, a.w + b.w};
  float ex[4];
  #pragma unroll
  for (int h = 0; h < 4; ++h) {
    float x = v[h];
    x = x > 0.f ? x : NEG_SLOPE * x;
    const float mx = key2f(emaxk[d * 4 + h]);
    ex[h] = __expf(x - mx);
    atomicAddF32(&denom[d * 4 + h], ex[h]);
  }
  ((float4*)exbuf)[e] = make_float4(ex[0], ex[1], ex[2], ex[3]);
}

// rst[dst] += feat[src] * a   (wave per edge, 256 floats per edge)
__global__ __launch_bounds__(256) void edge_msg(
    const int* __restrict__ src, const int* __restrict__ dst,
    const float* __restrict__ feat, const float* __restrict__ exbuf,
    const float* __restrict__ denom, float* __restrict__ rst, int E) {
  const int lane = threadIdx.x & 31;
  const int e = blockIdx.x * 8 + (threadIdx.x >> 5);
  if (e >= E) return;
  const int s = src[e], d = dst[e];
  const float4 ex = ((const float4*)exbuf)[e];
  const float4 dn = ((const float4*)denom)[d];
  const float c[4] = {ex.x / dn.x, ex.y / dn.y, ex.z / dn.z, ex.w / dn.w};
  const float* f = feat + (size_t)s * FDIM;
  float* o = rst + (size_t)d * FDIM;
  #pragma unroll
  for (int j = 0; j < 8; ++j) {
    const int i = lane + 32 * j;            // head = j>>1
    atomicAddF32(&o[i], f[i] * c[j >> 1]);
  }
}

// x = elu(x) in place
__global__ __launch_bounds__(256) void elu_kernel(float* __restrict__ x, int total) {
  const int i = blockIdx.x * blockDim.x + threadIdx.x;
  if (i < total) {
    const float v = x[i];
    x[i] = v > 0.f ? v : (__expf(v) - 1.f);
  }
}

// =====================================================================
static void run_layer(const float* hin, const float* Wm, const float* al,
                      const float* ar, const int* src, const int* dst,
                      int N, int E, float* feat, float* el, float* er,
                      unsigned* emaxk, float* denom, float* exbuf,
                      const float* resid, float* rst, hipStream_t stream) {
  dim3 ggrid((N + 63) / 64, FDIM / 64);
  gemm_f32_wmma<<<ggrid, 128, 0, stream>>>(hin, Wm, feat, N);
  node_attn<<<(N + 7) / 8, 256, 0, stream>>>(feat, al, ar, el, er, emaxk, denom, N);
  const int total4 = N * (FDIM / 4);
  init_acc<<<(total4 + 255) / 256, 256, 0, stream>>>(resid, rst, total4);
  edge_max<<<(E + 255) / 256, 256, 0, stream>>>(src, dst, el, er, emaxk, E);
  edge_sum<<<(E + 255) / 256, 256, 0, stream>>>(src, dst, el, er, emaxk, denom, exbuf, E);
  edge_msg<<<(E + 7) / 8, 256, 0, stream>>>(src, dst, feat, exbuf, denom, rst, E);
  elu_kernel<<<(N * FDIM + 255) / 256, 256, 0, stream>>>(rst, N * FDIM);
}

extern "C" void kernel_launch(void* const* d_in, const int* in_sizes, int n_in,
                              void* d_out, int out_size, void* d_ws, size_t ws_size,
                              hipStream_t stream) {
  const float* features = (const float*)d_in[0];
  const int*   src      = (const int*)d_in[1];
  const int*   dst      = (const int*)d_in[2];
  const float* W1       = (const float*)d_in[3];
  const float* al1      = (const float*)d_in[4];
  const float* ar1      = (const float*)d_in[5];
  const float* W2       = (const float*)d_in[6];
  const float* al2      = (const float*)d_in[7];
  const float* ar2      = (const float*)d_in[8];

  const int N = in_sizes[0] / FDIM;
  const int E = in_sizes[1];

  char* w = (char*)d_ws;
  float* feat = (float*)w;  w += (size_t)N * FDIM * sizeof(float);
  float* h1   = (float*)w;  w += (size_t)N * FDIM * sizeof(float);
  float* el   = (float*)w;  w += (size_t)N * HEADS * sizeof(float);
  float* er   = (float*)w;  w += (size_t)N * HEADS * sizeof(float);
  unsigned* emaxk = (unsigned*)w; w += (size_t)N * HEADS * sizeof(unsigned);
  float* denom = (float*)w; w += (size_t)N * HEADS * sizeof(float);
  float* exbuf = (float*)w; // E * 4 floats

  // layer 1: no residual, output -> h1 (in ws)
  run_layer(features, W1, al1, ar1, src, dst, N, E,
            feat, el, er, emaxk, denom, exbuf,
            /*resid=*/nullptr, /*rst=*/h1, stream);

  // layer 2: residual = h1, accumulate directly into d_out
  run_layer(h1, W2, al2, ar2, src, dst, N, E,
            feat, el, er, emaxk, denom, exbuf,
            /*resid=*/h1, /*rst=*/(float*)d_out, stream);
}